// MTGNNLayer_18811956756538
// MI455X (gfx1250) — compile-verified
//
#include <hip/hip_runtime.h>
#include <hip/hip_bf16.h>

typedef float v2f __attribute__((ext_vector_type(2)));
typedef float v8f __attribute__((ext_vector_type(8)));

#define ALPHA 0.05f
#define FDIM 4096
#define MROWS 2048   // B*L
#define NNODE 128
#define ELEMS 8388608ULL  // B*C*N*L

// ---------------------------------------------------------------------------
// Adjacency prep: Aoff = (A!=0) with zero diagonal; deg/dis for A and A^T
// ---------------------------------------------------------------------------
__global__ __launch_bounds__(128) void prep_adj(const float* __restrict__ A,
    float* __restrict__ Aoff, float* __restrict__ dis1, float* __restrict__ inv1,
    float* __restrict__ dis2, float* __restrict__ inv2)
{
    int t = threadIdx.x;
    float colsum = 0.f, rowsum = 0.f;
    for (int r = 0; r < 128; ++r) {
        float art = A[r * 128 + t];
        float v = (art != 0.f && r != t) ? 1.f : 0.f;
        Aoff[r * 128 + t] = v;
        colsum += v;
        float atr = A[t * 128 + r];
        rowsum += (atr != 0.f && r != t) ? 1.f : 0.f;
    }
    float d1 = 1.f + colsum;
    dis1[t] = rsqrtf(d1);
    inv1[t] = 1.f / d1;
    float d2 = 1.f + rowsum;
    dis2[t] = rsqrtf(d2);
    inv2[t] = 1.f / d2;
}

// ---------------------------------------------------------------------------
// Gated inception: X0 = tanh(conv_f(x)) * sigmoid(conv_g(x)); dilation 2
// One block per (b, node); 256 threads: 32 channels x 8 L-groups of 16
// ---------------------------------------------------------------------------
__global__ __launch_bounds__(256) void inception_kernel(
    const float* __restrict__ x,
    const float* __restrict__ fw2, const float* __restrict__ fb2,
    const float* __restrict__ fw3, const float* __restrict__ fb3,
    const float* __restrict__ fw6, const float* __restrict__ fb6,
    const float* __restrict__ fw7, const float* __restrict__ fb7,
    const float* __restrict__ gw2, const float* __restrict__ gb2,
    const float* __restrict__ gw3, const float* __restrict__ gb3,
    const float* __restrict__ gw6, const float* __restrict__ gb6,
    const float* __restrict__ gw7, const float* __restrict__ gb7,
    float* __restrict__ X0)
{
    __shared__ float xs[32][128];
    __shared__ float fws[4608], gws[4608];
    __shared__ float fbs[32], gbsS[32];
    const int ksz[4] = {2, 3, 6, 7};
    const int woff[4] = {0, 512, 1280, 2816};
    int tid = threadIdx.x;
    int b = blockIdx.x >> 7;
    int xn = blockIdx.x & 127;

    for (int i = tid; i < 32 * 128; i += 256) {
        int ci = i >> 7, l = i & 127;
        xs[ci][l] = x[(((size_t)b * 32 + ci) * 128 + xn) * 128 + l];
    }
    const float* fsrc[4] = {fw2, fw3, fw6, fw7};
    const float* gsrc[4] = {gw2, gw3, gw6, gw7};
    const float* fbsrc[4] = {fb2, fb3, fb6, fb7};
    const float* gbsrc[4] = {gb2, gb3, gb6, gb7};
    for (int br = 0; br < 4; ++br) {
        int nelem = 256 * ksz[br];   // 8*32*k
        for (int i = tid; i < nelem; i += 256) {
            fws[woff[br] + i] = fsrc[br][i];
            gws[woff[br] + i] = gsrc[br][i];
        }
    }
    if (tid < 32) {
        int br = tid >> 3, oc = tid & 7;
        fbs[tid] = fbsrc[br][oc];
        gbsS[tid] = gbsrc[br][oc];
    }
    __syncthreads();

    int c = tid >> 3;
    int br = c >> 3, oc = c & 7;
    int k = ksz[br];
    int lo = k - 1;   // lo = (2*(k-1))/2
    const float* wf = &fws[woff[br] + oc * 32 * k];
    const float* wg = &gws[woff[br] + oc * 32 * k];
    float bf = fbs[c], bg = gbsS[c];
    size_t obase = (((size_t)b * 32 + c) * 128 + xn) * 128;
    for (int i = 0; i < 16; ++i) {
        int l = (tid & 7) * 16 + i;
        float sf = bf, sg = bg;
        for (int ci = 0; ci < 32; ++ci) {
            const float* row = xs[ci];
            for (int j = 0; j < k; ++j) {
                int pos = l + 2 * j - lo;
                if (pos >= 0 && pos < 128) {
                    float v = row[pos];
                    sf += wf[ci * k + j] * v;
                    sg += wg[ci * k + j] * v;
                }
            }
        }
        float t = tanhf(sf);
        float s = 1.f / (1.f + expf(-sg));
        X0[obase + l] = t * s;
    }
}

// ---------------------------------------------------------------------------
// Skip path: out = X0 einsum skip_w + skip_b + x_skip
// ---------------------------------------------------------------------------
__global__ __launch_bounds__(128) void skip_kernel(
    const float* __restrict__ X0, const float* __restrict__ skip_w,
    const float* __restrict__ skip_b, const float* __restrict__ xskip_in,
    float* __restrict__ out)
{
    __shared__ float sw[64 * 32];
    __shared__ float sb[64];
    int tid = threadIdx.x;
    int b = blockIdx.x >> 7, xn = blockIdx.x & 127;
    for (int i = tid; i < 2048; i += 128) sw[i] = skip_w[i];
    if (tid < 64) sb[tid] = skip_b[tid];
    __syncthreads();
    int l = tid;
    float xv[32];
    for (int c = 0; c < 32; ++c)
        xv[c] = X0[(((size_t)b * 32 + c) * 128 + xn) * 128 + l];
    for (int o = 0; o < 64; ++o) {
        float s = sb[o];
        for (int c = 0; c < 32; ++c) s += sw[o * 32 + c] * xv[c];
        size_t oi = (((size_t)b * 64 + o) * 128 + xn) * 128 + l;
        out[oi] = s + xskip_in[oi];
    }
}

// ---------------------------------------------------------------------------
// WMMA f32 GEMM: Out(2048x4096) = Mmat(2048x4096) @ Wmat(4096x4096)^T
// 64x64 tile / block, 4 waves, each wave 32x32 (2x2 WMMA tiles), K-chunk 16
// ---------------------------------------------------------------------------
__global__ __launch_bounds__(128) void gemm_xw(
    const float* __restrict__ Mmat, const float* __restrict__ Wmat,
    float* __restrict__ Out)
{
    __shared__ float Atile[64][17];
    __shared__ float Btile[16][65];
    int tid = threadIdx.x;
    int lane = tid & 31;
    int wave = tid >> 5;
    int r0 = blockIdx.y * 64;
    int c0 = blockIdx.x * 64;
    int wm = (wave >> 1) * 32;
    int wn = (wave & 1) * 32;
    int lm = lane & 15;
    int kh = lane >> 4;   // 0: K pair {0,1}, 1: K pair {2,3}
    v8f acc[2][2] = {};

    for (int k0 = 0; k0 < FDIM; k0 += 16) {
        {
            int m = tid >> 1;
            int half = (tid & 1) * 8;
            const float* src = Mmat + (size_t)(r0 + m) * FDIM + k0 + half;
            const float* wsrc = Wmat + (size_t)(c0 + m) * FDIM + k0 + half;
            #pragma unroll
            for (int i = 0; i < 8; ++i) {
                Atile[m][half + i] = src[i];
                Btile[half + i][m] = wsrc[i];
            }
        }
        __syncthreads();
        #pragma unroll
        for (int kk = 0; kk < 16; kk += 4) {
            v2f afr[2], bfr[2];
            #pragma unroll
            for (int mt = 0; mt < 2; ++mt) {
                int mrow = wm + mt * 16 + lm;
                afr[mt].x = Atile[mrow][kk + kh * 2 + 0];
                afr[mt].y = Atile[mrow][kk + kh * 2 + 1];
            }
            #pragma unroll
            for (int nt = 0; nt < 2; ++nt) {
                int ncol = wn + nt * 16 + lm;
                bfr[nt].x = Btile[kk + kh * 2 + 0][ncol];
                bfr[nt].y = Btile[kk + kh * 2 + 1][ncol];
            }
            #pragma unroll
            for (int mt = 0; mt < 2; ++mt)
                #pragma unroll
                for (int nt = 0; nt < 2; ++nt)
                    acc[mt][nt] = __builtin_amdgcn_wmma_f32_16x16x4_f32(
                        false, afr[mt], false, bfr[nt],
                        (short)0, acc[mt][nt], false, false);
        }
        __syncthreads();
    }
    #pragma unroll
    for (int mt = 0; mt < 2; ++mt)
        #pragma unroll
        for (int nt = 0; nt < 2; ++nt) {
            int col = c0 + wn + nt * 16 + lm;
            #pragma unroll
            for (int j = 0; j < 8; ++j) {
                int row = r0 + wm + mt * 16 + j + kh * 8;
                Out[(size_t)row * FDIM + col] = acc[mt][nt][j];
            }
        }
}

// ---------------------------------------------------------------------------
// Graph propagation: P[r,f] = sum_s Aeff(s,r) * dis[s] * XW[s,f], r < 128
// trans==0: Aeff(s,r)=Aoff[s][r] (mp1);  trans==1: Aeff(s,r)=Aoff[r][s] (mp2)
// ---------------------------------------------------------------------------
__global__ __launch_bounds__(256) void graph_prop(
    const float* __restrict__ Aoff, const float* __restrict__ dis,
    const float* __restrict__ XW, float* __restrict__ P, int trans)
{
    __shared__ float coef[32][128];
    int tid = threadIdx.x;
    int f0 = blockIdx.x * 256;
    int r0 = blockIdx.y * 32;
    for (int i = tid; i < 32 * 128; i += 256) {
        int ri = i >> 7, s = i & 127;
        float a = trans ? Aoff[(r0 + ri) * 128 + s] : Aoff[s * 128 + (r0 + ri)];
        coef[ri][s] = a * dis[s];
    }
    __syncthreads();
    int f = f0 + tid;
    float acc[32];
    #pragma unroll
    for (int ri = 0; ri < 32; ++ri) acc[ri] = 0.f;
    for (int s = 0; s < 128; ++s) {
        float v = XW[(size_t)s * FDIM + f];
        #pragma unroll
        for (int ri = 0; ri < 32; ++ri) acc[ri] += coef[ri][s] * v;
    }
    for (int ri = 0; ri < 32; ++ri) P[(size_t)(r0 + ri) * FDIM + f] = acc[ri];
}

// ---------------------------------------------------------------------------
// H = ALPHA*X0 + (XW/deg + [r<128] dis[r]*P + bias)
// ---------------------------------------------------------------------------
__global__ __launch_bounds__(256) void hop_update(
    const float* __restrict__ X0, const float* __restrict__ XW,
    const float* __restrict__ P, const float* __restrict__ invdeg,
    const float* __restrict__ dis, const float* __restrict__ bias,
    float* __restrict__ H)
{
    size_t idx = (size_t)blockIdx.x * 256 + threadIdx.x;
    int r = (int)(idx >> 12);
    int f = (int)(idx & 4095);
    float v = XW[idx];
    float o;
    if (r < 128) o = v * invdeg[r] + dis[r] * P[(size_t)r * FDIM + f];
    else         o = v;
    H[idx] = ALPHA * X0[idx] + o + bias[f];
}

// ---------------------------------------------------------------------------
// Mix: ACC += einsum over channel concat [X0, HA, HB] with mw (32x96) + mb
// ---------------------------------------------------------------------------
__global__ __launch_bounds__(128) void mix_kernel(
    const float* __restrict__ X0, const float* __restrict__ HA,
    const float* __restrict__ HB, const float* __restrict__ mw,
    const float* __restrict__ mb, float* __restrict__ ACCb)
{
    __shared__ float mwS[32 * 96];
    __shared__ float mbS[32];
    int tid = threadIdx.x;
    int b = blockIdx.x >> 7, xn = blockIdx.x & 127;
    for (int i = tid; i < 32 * 96; i += 128) mwS[i] = mw[i];
    if (tid < 32) mbS[tid] = mb[tid];
    __syncthreads();
    int l = tid;
    float accv[32];
    #pragma unroll
    for (int o = 0; o < 32; ++o) accv[o] = mbS[o];
    for (int c = 0; c < 32; ++c) {
        size_t base = (((size_t)b * 32 + c) * 128 + xn) * 128 + l;
        float v0 = X0[base], v1 = HA[base], v2 = HB[base];
        #pragma unroll
        for (int o = 0; o < 32; ++o)
            accv[o] += mwS[o * 96 + c] * v0 + mwS[o * 96 + 32 + c] * v1
                     + mwS[o * 96 + 64 + c] * v2;
    }
    for (int o = 0; o < 32; ++o) {
        size_t oi = (((size_t)b * 32 + o) * 128 + xn) * 128 + l;
        ACCb[oi] += accv[o];
    }
}

// ---------------------------------------------------------------------------
// LayerNorm: deterministic two-stage reduction + apply
// ---------------------------------------------------------------------------
__global__ __launch_bounds__(256) void ln_partials_kernel(
    const float* __restrict__ ACCb, float* __restrict__ partials)
{
    __shared__ float s1[256], s2[256];
    int tid = threadIdx.x;
    int b = blockIdx.y, chunk = blockIdx.x;
    size_t base = (size_t)b * 524288 + (size_t)chunk * 8192;
    float sum = 0.f, sq = 0.f;
    for (int i = tid; i < 8192; i += 256) {
        float v = ACCb[base + i];
        sum += v; sq += v * v;
    }
    s1[tid] = sum; s2[tid] = sq;
    __syncthreads();
    for (int st = 128; st > 0; st >>= 1) {
        if (tid < st) { s1[tid] += s1[tid + st]; s2[tid] += s2[tid + st]; }
        __syncthreads();
    }
    if (tid == 0) {
        partials[(b * 64 + chunk) * 2 + 0] = s1[0];
        partials[(b * 64 + chunk) * 2 + 1] = s2[0];
    }
}

__global__ __launch_bounds__(64) void ln_stats_kernel(
    const float* __restrict__ partials, float* __restrict__ stats)
{
    __shared__ float s1[64], s2[64];
    int tid = threadIdx.x, b = blockIdx.x;
    s1[tid] = partials[(b * 64 + tid) * 2 + 0];
    s2[tid] = partials[(b * 64 + tid) * 2 + 1];
    __syncthreads();
    for (int st = 32; st > 0; st >>= 1) {
        if (tid < st) { s1[tid] += s1[tid + st]; s2[tid] += s2[tid + st]; }
        __syncthreads();
    }
    if (tid == 0) {
        float mean = s1[0] / 524288.f;
        float var = s2[0] / 524288.f - mean * mean;
        stats[b * 2 + 0] = mean;
        stats[b * 2 + 1] = rsqrtf(var + 1e-5f);
    }
}

__global__ __launch_bounds__(256) void ln_apply_kernel(
    const float* __restrict__ ACCb, const float* __restrict__ stats,
    const float* __restrict__ lnw, const float* __restrict__ lnb,
    const int* __restrict__ idxv, float* __restrict__ outx)
{
    size_t i = (size_t)blockIdx.x * 256 + threadIdx.x;
    int b = (int)(i >> 19);
    int rem = (int)(i & 524287);
    int c = rem >> 14;
    int xn = (rem >> 7) & 127;
    int l = rem & 127;
    int wi = (c * 128 + idxv[xn]) * 128 + l;
    float mean = stats[b * 2], rstd = stats[b * 2 + 1];
    outx[i] = (ACCb[i] - mean) * rstd * lnw[wi] + lnb[wi];
}

// ---------------------------------------------------------------------------
extern "C" void kernel_launch(void* const* d_in, const int* in_sizes, int n_in,
                              void* d_out, int out_size, void* d_ws, size_t ws_size,
                              hipStream_t stream)
{
    const float* x       = (const float*)d_in[0];
    const float* xskip   = (const float*)d_in[1];
    const float* A       = (const float*)d_in[2];
    const int*   idxv    = (const int*)d_in[3];
    const float* fw2 = (const float*)d_in[4],  *fb2 = (const float*)d_in[5];
    const float* fw3 = (const float*)d_in[6],  *fb3 = (const float*)d_in[7];
    const float* fw6 = (const float*)d_in[8],  *fb6 = (const float*)d_in[9];
    const float* fw7 = (const float*)d_in[10], *fb7 = (const float*)d_in[11];
    const float* gw2 = (const float*)d_in[12], *gb2 = (const float*)d_in[13];
    const float* gw3 = (const float*)d_in[14], *gb3 = (const float*)d_in[15];
    const float* gw6 = (const float*)d_in[16], *gb6 = (const float*)d_in[17];
    const float* gw7 = (const float*)d_in[18], *gb7 = (const float*)d_in[19];
    const float* skip_w = (const float*)d_in[20];
    const float* skip_b = (const float*)d_in[21];
    const float* mp1_gw = (const float*)d_in[22];
    const float* mp1_gb = (const float*)d_in[23];
    const float* mp1_mw = (const float*)d_in[24];
    const float* mp1_mb = (const float*)d_in[25];
    const float* mp2_gw = (const float*)d_in[26];
    const float* mp2_gb = (const float*)d_in[27];
    const float* mp2_mw = (const float*)d_in[28];
    const float* mp2_mb = (const float*)d_in[29];
    const float* lnw    = (const float*)d_in[30];
    const float* lnb    = (const float*)d_in[31];

    float* ws = (float*)d_ws;
    const size_t SZ = ELEMS;
    float* X0   = ws;
    float* HA   = ws + SZ;
    float* HB   = ws + 2 * SZ;
    float* XWb  = ws + 3 * SZ;
    float* ACCb = ws + 4 * SZ;
    float* Pb   = ws + 5 * SZ;              // 524288
    float* Aoffb = Pb + 524288;             // 16384
    float* dis1 = Aoffb + 16384;
    float* inv1 = dis1 + 128;
    float* dis2 = inv1 + 128;
    float* inv2 = dis2 + 128;
    float* partials = inv2 + 128;           // 2048
    float* stats = partials + 2048;         // 32

    float* out_x = (float*)d_out;
    float* out_skip = out_x + SZ;

    prep_adj<<<1, 128, 0, stream>>>(A, Aoffb, dis1, inv1, dis2, inv2);
    inception_kernel<<<2048, 256, 0, stream>>>(x,
        fw2, fb2, fw3, fb3, fw6, fb6, fw7, fb7,
        gw2, gb2, gw3, gb3, gw6, gb6, gw7, gb7, X0);
    skip_kernel<<<2048, 128, 0, stream>>>(X0, skip_w, skip_b, xskip, out_skip);
    hipMemcpyAsync(ACCb, x, SZ * sizeof(float), hipMemcpyDeviceToDevice, stream);

    for (int mp = 0; mp < 2; ++mp) {
        const float* gw  = mp == 0 ? mp1_gw : mp2_gw;
        const float* gb  = mp == 0 ? mp1_gb : mp2_gb;
        const float* mwp = mp == 0 ? mp1_mw : mp2_mw;
        const float* mbp = mp == 0 ? mp1_mb : mp2_mb;
        const float* dis = mp == 0 ? dis1 : dis2;
        const float* inv = mp == 0 ? inv1 : inv2;
        // hop 0
        gemm_xw<<<dim3(FDIM / 64, MROWS / 64), 128, 0, stream>>>(X0, gw, XWb);
        graph_prop<<<dim3(16, 4), 256, 0, stream>>>(Aoffb, dis, XWb, Pb, mp);
        hop_update<<<32768, 256, 0, stream>>>(X0, XWb, Pb, inv, dis, gb, HA);
        // hop 1
        gemm_xw<<<dim3(FDIM / 64, MROWS / 64), 128, 0, stream>>>(
            HA, gw + (size_t)FDIM * FDIM, XWb);
        graph_prop<<<dim3(16, 4), 256, 0, stream>>>(Aoffb, dis, XWb, Pb, mp);
        hop_update<<<32768, 256, 0, stream>>>(X0, XWb, Pb, inv, dis, gb + FDIM, HB);
        // channel mix, accumulate into pre-LN buffer
        mix_kernel<<<2048, 128, 0, stream>>>(X0, HA, HB, mwp, mbp, ACCb);
    }

    ln_partials_kernel<<<dim3(64, 16), 256, 0, stream>>>(ACCb, partials);
    ln_stats_kernel<<<16, 64, 0, stream>>>(partials, stats);
    ln_apply_kernel<<<32768, 256, 0, stream>>>(ACCb, stats, lnw, lnb, idxv, out_x);
}